// AbsDiff_46583215292572
// MI455X (gfx1250) — compile-verified
//
#include <hip/hip_runtime.h>
#include <cstdint>

// Cost volume: out[d*C+ch, h, x] = |left[ch,h,x+d] - right[ch,h,x]|, 0 for x >= W-d
// Shapes: left/right [1,16,256,512] f32, out [1,1024,256,512] f32 (512 MB).
// Store-bandwidth bound: floor ~ 512MB / 23.3TB/s ~ 22us on MI455X.

#define W 512
#define H 256
#define C 16
#define DMAX 64

typedef float v4f __attribute__((ext_vector_type(4)));

__global__ __launch_bounds__(256) void cost_volume_kernel(
    const float* __restrict__ left,
    const float* __restrict__ right,
    float* __restrict__ out)
{
    // Left row staged with a 128-float zero tail so the sliding disparity
    // window (up to index x+d+5 = 576) stays inside the array.
    __shared__ alignas(16) float sL[640];
    __shared__ alignas(16) float sR[512];

    const int tid = threadIdx.x;
    const int bid = blockIdx.x;         // 0 .. C*H-1
    const int ch  = bid & (C - 1);
    const int h   = bid >> 4;

    const float* lrow = left  + ((size_t)ch * H + h) * W;
    const float* rrow = right + ((size_t)ch * H + h) * W;

#if defined(__HIP_DEVICE_COMPILE__)
    if (tid < 128) {
        // CDNA5 async copy engine: per-lane 16B global->LDS, tracked on ASYNCcnt.
        // GVS mode: mem_addr = SADDR(64b, SGPR pair) + VADDR(32b byte offset).
        unsigned q16  = (unsigned)tid * 16u;
        unsigned lofs = (unsigned)(uintptr_t)(&sL[tid * 4]);
        unsigned rofs = (unsigned)(uintptr_t)(&sR[tid * 4]);
        asm volatile("global_load_async_to_lds_b128 %0, %1, %2"
                     :: "v"(lofs), "v"(q16), "s"(lrow) : "memory");
        asm volatile("global_load_async_to_lds_b128 %0, %1, %2"
                     :: "v"(rofs), "v"(q16), "s"(rrow) : "memory");
    } else {
        sL[512 + (tid - 128)] = 0.0f;   // zero the tail (values masked anyway)
    }
    __builtin_amdgcn_s_wait_asynccnt(0);
#else
    // Host-pass placeholder (never executed; device pass uses the asm above).
    if (tid < 128) {
        for (int j = 0; j < 4; ++j) {
            sL[tid * 4 + j] = lrow[tid * 4 + j];
            sR[tid * 4 + j] = rrow[tid * 4 + j];
        }
    } else {
        sL[512 + (tid - 128)] = 0.0f;
    }
#endif
    __syncthreads();

    // Thread -> (x..x+3, disparity parity). 256 threads = 128 x-quads x 2 parities.
    const int x     = (tid & 127) << 2;
    const int dbase = tid >> 7;           // 0 or 1

    const v4f r = *(const v4f*)(&sR[x]);  // aligned ds_load_b128, reused for all d

    // Rolling 4-float window of the left row starting at x + dbase; the window
    // slides by 2 floats per iteration (d += 2), so only 2 new LDS loads/iter.
    const int start = x + dbase;
    float a0 = sL[start + 0];
    float a1 = sL[start + 1];
    float a2 = sL[start + 2];
    float a3 = sL[start + 3];

    float* obase = out + (((size_t)(dbase * C + ch) * H + h) * W) + x;
    const size_t dstep = (size_t)2 * C * H * W;   // advance d by 2

    #pragma unroll 4
    for (int di = 0; di < 32; ++di) {
        const int d   = 2 * di + dbase;
        const int lim = W - d;            // valid iff x + j < lim
        v4f v;
        v.x = (x + 0 < lim) ? __builtin_fabsf(a0 - r.x) : 0.0f;
        v.y = (x + 1 < lim) ? __builtin_fabsf(a1 - r.y) : 0.0f;
        v.z = (x + 2 < lim) ? __builtin_fabsf(a2 - r.z) : 0.0f;
        v.w = (x + 3 < lim) ? __builtin_fabsf(a3 - r.w) : 0.0f;
        // 512MB streaming output (2.7x L2): non-temporal b128 store.
        __builtin_nontemporal_store(v, (v4f*)obase);

        a0 = a2;
        a1 = a3;
        a2 = sL[start + 2 * di + 4];      // max index 576 < 640
        a3 = sL[start + 2 * di + 5];
        obase += dstep;
    }
}

extern "C" void kernel_launch(void* const* d_in, const int* in_sizes, int n_in,
                              void* d_out, int out_size, void* d_ws, size_t ws_size,
                              hipStream_t stream) {
    (void)in_sizes; (void)n_in; (void)out_size; (void)d_ws; (void)ws_size;
    const float* left  = (const float*)d_in[0];
    const float* right = (const float*)d_in[1];
    float* out = (float*)d_out;

    dim3 grid(C * H);   // 4096 blocks, one per (channel, row)
    dim3 block(256);    // 8 wave32 waves
    cost_volume_kernel<<<grid, block, 0, stream>>>(left, right, out);
}